// HashTable_31078383354737
// MI455X (gfx1250) — compile-verified
//
#include <hip/hip_runtime.h>

#define VOX        0.01f
#define TABLE_SIZE 4194304        // 2^22 rows
#define TABLE_MASK (TABLE_SIZE - 1)
#define FDIM       16             // floats per row (64 bytes)

// 1 / (sqrt(3) * VOXEL_SIZE)
#define INV_DIAG 57.735026918962575f

typedef float v4f __attribute__((ext_vector_type(4)));

__global__ __launch_bounds__(256) void hashgrid_interp_kernel(
    const float* __restrict__ coords,   // (NPTS, 3)
    const float* __restrict__ table,    // (TABLE_SIZE, 16)
    float* __restrict__ out,            // (NPTS, 16)
    unsigned int npts)
{
    const unsigned int idx = blockIdx.x * blockDim.x + threadIdx.x;
    if (idx >= npts) return;

    // ---- load point (single global_load_b96) ----
    const float cx = coords[3u * idx + 0u];
    const float cy = coords[3u * idx + 1u];
    const float cz = coords[3u * idx + 2u];

    // base_coords = float(int(floor(c / VOX))) * VOX
    // NOTE: keep IEEE division here — floor() boundary sensitivity means a
    // 1-ulp deviation could pick a different voxel (wrong hash bucket).
    const float bx = floorf(cx / VOX) * VOX;
    const float by = floorf(cy / VOX) * VOX;
    const float bz = floorf(cz / VOX) * VOX;

    const float p0 = 73856093.0f, p1 = 19349663.0f, p2 = 83492791.0f;

    // corner offsets (CORNERS * VOXEL_SIZE), same order as reference
    const float OX[8] = {0.f, VOX, VOX, 0.f, 0.f, VOX, VOX, 0.f};
    const float OY[8] = {0.f, 0.f, VOX, VOX, 0.f, 0.f, VOX, VOX};
    const float OZ[8] = {0.f, 0.f, 0.f, 0.f, VOX, VOX, VOX, VOX};

    // ---- phase 1: all 8 hashes + weights (pure VALU, no memory deps) ----
    unsigned int row[8];
    float        wgt[8];
#pragma unroll
    for (int k = 0; k < 8; ++k) {
        const float vx = bx + OX[k];
        const float vy = by + OY[k];
        const float vz = bz + OZ[k];

        const float dx = cx - vx, dy = cy - vy, dz = cz - vz;
        // raw v_sqrt_f32 + reciprocal-multiply: weight-only precision,
        // <=1 ulp off the reference's sqrt/div — invisible in the output sum.
        wgt[k] = __builtin_amdgcn_sqrtf(dx * dx + dy * dy + dz * dz) * INV_DIAG;

        // float-math hash, jnp.mod(x, y) = x - floor(x/y)*y
        const float h  = vx * p0 + vy * p1 + vz * p2;
        const float q  = floorf(h * (1.0f / (float)TABLE_SIZE));
        const float r  = h - q * (float)TABLE_SIZE;
        row[k] = ((unsigned int)(int)r) & TABLE_MASK;   // 2^22 -> safe wrap
    }

    // ---- phase 2: 32x global_load_b128 gathers, fully unrolled for MLP ----
    const v4f* __restrict__ t4 = (const v4f*)table;
    v4f a0 = {0.f, 0.f, 0.f, 0.f};
    v4f a1 = {0.f, 0.f, 0.f, 0.f};
    v4f a2 = {0.f, 0.f, 0.f, 0.f};
    v4f a3 = {0.f, 0.f, 0.f, 0.f};

#pragma unroll
    for (int k = 0; k < 8; ++k) {
        const size_t r4 = (size_t)row[k] * (FDIM / 4);  // v4f index of row
        const v4f f0 = t4[r4 + 0];
        const v4f f1 = t4[r4 + 1];
        const v4f f2 = t4[r4 + 2];
        const v4f f3 = t4[r4 + 3];
        const float w = wgt[k];
        a0 += w * f0;
        a1 += w * f1;
        a2 += w * f2;
        a3 += w * f3;
    }

    // ---- streaming output: non-temporal b128 stores (don't pollute L2) ----
    v4f* __restrict__ o4 = (v4f*)out;
    const size_t ob = (size_t)idx * (FDIM / 4);
    __builtin_nontemporal_store(a0, &o4[ob + 0]);
    __builtin_nontemporal_store(a1, &o4[ob + 1]);
    __builtin_nontemporal_store(a2, &o4[ob + 2]);
    __builtin_nontemporal_store(a3, &o4[ob + 3]);
}

extern "C" void kernel_launch(void* const* d_in, const int* in_sizes, int n_in,
                              void* d_out, int out_size, void* d_ws, size_t ws_size,
                              hipStream_t stream)
{
    const float* coords = (const float*)d_in[0];   // (4096, 128, 3) f32
    const float* table  = (const float*)d_in[1];   // (2^22, 16) f32
    float*       out    = (float*)d_out;           // (4096, 128, 16) f32

    const unsigned int npts = (unsigned int)(in_sizes[0] / 3);  // 524288
    const int block = 256;                                      // 8 wave32 waves
    const int grid  = (int)((npts + block - 1) / block);

    hashgrid_interp_kernel<<<grid, block, 0, stream>>>(coords, table, out, npts);
}